// GraphKnowledgeAggregation_71588514890457
// MI455X (gfx1250) — compile-verified
//
#include <hip/hip_runtime.h>
#include <stdint.h>

// Problem constants (from reference)
#define N_NODES   10000
#define FEAT      128
#define N_EDGES   640000
#define AGG       0.3f
#define EPSV      1e-8f

// Tiling
#define BIN_NODES 64                                      // nodes per bin
#define NBINS     ((N_NODES + BIN_NODES - 1) / BIN_NODES) // 157
#define BLOCK     1024                                    // 32 waves (wave32)
#define TILE      1024                                    // edges per staged tile
#define NTILES    (N_EDGES / TILE)                        // 625, exact
#define SLICES    5                                       // edge-stream slices
#define TILES_PER_SLICE (NTILES / SLICES)                 // 125, exact

#define WS_ROWS   (NBINS * BIN_NODES)                     // 10048 (>= N_NODES)
#define WS_FLOATS ((size_t)WS_ROWS * (FEAT + 1))          // accum + counts

// ---- CDNA5 helpers -------------------------------------------------------

// Low 32 bits of a generic pointer to __shared__ = LDS byte address.
__device__ __forceinline__ uint32_t lds_addr_of(const void* p) {
  return (uint32_t)(uintptr_t)p;
}

// Async global->LDS copies (ASYNCcnt path).
__device__ __forceinline__ void async_ld_b128(uint32_t lds, uint64_t gaddr) {
  asm volatile("global_load_async_to_lds_b128 %0, %1, off"
               :: "v"(lds), "v"(gaddr) : "memory");
}
__device__ __forceinline__ void async_ld_b64(uint32_t lds, uint64_t gaddr) {
  asm volatile("global_load_async_to_lds_b64 %0, %1, off"
               :: "v"(lds), "v"(gaddr) : "memory");
}
__device__ __forceinline__ void async_ld_b32(uint32_t lds, uint64_t gaddr) {
  asm volatile("global_load_async_to_lds_b32 %0, %1, off"
               :: "v"(lds), "v"(gaddr) : "memory");
}
__device__ __forceinline__ void wait_async0() {
#if __has_builtin(__builtin_amdgcn_s_wait_asynccnt)
  __builtin_amdgcn_s_wait_asynccnt(0);
#else
  asm volatile("s_wait_asynccnt 0" ::: "memory");
#endif
}

__device__ __forceinline__ unsigned ballot32(bool p) {
#if __has_builtin(__builtin_amdgcn_ballot_w32)
  return __builtin_amdgcn_ballot_w32(p);
#else
  return (unsigned)__ballot((int)p);
#endif
}

// LDS float atomic add -> ds_add_f32
__device__ __forceinline__ void lds_fadd(float* p, float v) {
  __hip_atomic_fetch_add(p, v, __ATOMIC_RELAXED, __HIP_MEMORY_SCOPE_WORKGROUP);
}
// Global float atomic add (relaxed, device scope) -> global_atomic_add_f32
__device__ __forceinline__ void glb_fadd(float* p, float v) {
  __hip_atomic_fetch_add(p, v, __ATOMIC_RELAXED, __HIP_MEMORY_SCOPE_AGENT);
}

// ---- Shared tile machinery (common to both paths) ------------------------

#define DECLARE_LDS()                                        \
  __shared__ float s_acc[BIN_NODES * FEAT];  /* 32 KB */     \
  __shared__ float s_cnt[BIN_NODES];                         \
  __shared__ __align__(16) int2  s_pairs[2][TILE]; /* 16KB */\
  __shared__ __align__(16) float s_w[2][TILE]      /* 8 KB */

// Issue async stage of one tile (edgeBase = first edge index, full tile).
// b128 packing: threads 0..511 copy 2 edge pairs each, 0..255 copy 4 weights.
__device__ __forceinline__ void stage_tile(int buf, int edgeBase, int tid,
                                           const int* edges, const float* ew,
                                           int2 (*s_pairs)[TILE],
                                           float (*s_w)[TILE]) {
  if (tid < TILE / 2) {
    async_ld_b128(lds_addr_of(&s_pairs[buf][2 * tid]),
                  (uint64_t)(uintptr_t)(edges + 2 * (edgeBase + 2 * tid)));
  }
  if (tid < TILE / 4) {
    async_ld_b128(lds_addr_of(&s_w[buf][4 * tid]),
                  (uint64_t)(uintptr_t)(ew + edgeBase + 4 * tid));
  }
}

// Filter + scatter one tile: ballot-compact matches, wave-cooperative
// float4 gather + 4x ds_add_f32 per matched edge.
__device__ __forceinline__ void process_tile(int cur, int tid, int lane,
                                             int nodeLo,
                                             const float4* __restrict__ feat4,
                                             int2 (*s_pairs)[TILE],
                                             float (*s_w)[TILE],
                                             float* s_acc, float* s_cnt) {
  const int2  pr  = s_pairs[cur][tid];
  const float w   = s_w[cur][tid];
  const int   tl  = pr.y - nodeLo;
  const bool  hit = (unsigned)tl < (unsigned)BIN_NODES;
  if (hit) lds_fadd(&s_cnt[tl], w);

  unsigned m = ballot32(hit);
  const int chunk = tid & ~31;                 // this wave's 32-edge base
  while (m) {
    const int b = __builtin_ctz(m);
    m &= m - 1;
    const int   j   = chunk + b;
    const int2  p2  = s_pairs[cur][j];         // uniform LDS read (broadcast)
    const float w2  = s_w[cur][j];
    const int   row = (p2.y - nodeLo) * FEAT + lane * 4;
    const float4 f4 = feat4[p2.x * (FEAT / 4) + lane];
    lds_fadd(&s_acc[row + 0], f4.x * w2);
    lds_fadd(&s_acc[row + 1], f4.y * w2);
    lds_fadd(&s_acc[row + 2], f4.z * w2);
    lds_fadd(&s_acc[row + 3], f4.w * w2);
  }
}

// ---- Path A: sliced pipeline (needs d_ws) --------------------------------

__global__ void zero_ws_kernel(float* p, int n) {
  const int i = blockIdx.x * blockDim.x + threadIdx.x;
  if (i < n) p[i] = 0.0f;
}

__global__ __launch_bounds__(BLOCK)
void agg_accum_kernel(const float* __restrict__ features,
                      const float* __restrict__ ew,
                      const int*   __restrict__ edges,
                      float*       __restrict__ ws_acc,   // [WS_ROWS][FEAT]
                      float*       __restrict__ ws_cnt) { // [WS_ROWS]
  DECLARE_LDS();
  const int tid    = threadIdx.x;
  const int lane   = tid & 31;
  const int bin    = blockIdx.x;
  const int slice  = blockIdx.y;
  const int nodeLo = bin * BIN_NODES;
  const int tile0  = slice * TILES_PER_SLICE;

  for (int i = tid; i < BIN_NODES * FEAT; i += BLOCK) s_acc[i] = 0.0f;
  if (tid < BIN_NODES) s_cnt[tid] = 0.0f;

  stage_tile(0, tile0 * TILE, tid, edges, ew, s_pairs, s_w);
  __syncthreads();

  const float4* __restrict__ feat4 = (const float4*)features;

  for (int t = 0; t < TILES_PER_SLICE; ++t) {
    const int cur = t & 1;
    wait_async0();
    __syncthreads();
    if (t + 1 < TILES_PER_SLICE)
      stage_tile(1 - cur, (tile0 + t + 1) * TILE, tid, edges, ew, s_pairs, s_w);
    process_tile(cur, tid, lane, nodeLo, feat4, s_pairs, s_w, s_acc, s_cnt);
    // buf[cur] is only overwritten after the next iteration's top barrier.
  }

  __syncthreads();
  // Flush partial bin to global accumulator.
  for (int i = tid; i < BIN_NODES * FEAT; i += BLOCK)
    glb_fadd(&ws_acc[nodeLo * FEAT + i], s_acc[i]);
  if (tid < BIN_NODES)
    glb_fadd(&ws_cnt[nodeLo + tid], s_cnt[tid]);
}

__global__ void finalize_kernel(const float* __restrict__ features,
                                const float* __restrict__ ws_acc,
                                const float* __restrict__ ws_cnt,
                                float*       __restrict__ out) {
  const int i = blockIdx.x * blockDim.x + threadIdx.x;
  if (i >= N_NODES * FEAT) return;
  const int   g   = i >> 7;              // FEAT == 128
  const float c0  = ws_cnt[g];
  const float c   = fmaxf(c0, EPSV);
  const float msk = (c0 > EPSV) ? 1.0f : 0.0f;
  const float am  = AGG * msk;
  out[i] = features[i] * (1.0f - am) + ws_acc[i] * (am / c);
}

// ---- Path B: self-contained single kernel (no workspace) -----------------

__global__ __launch_bounds__(BLOCK)
void agg_single_kernel(const float* __restrict__ features,
                       const float* __restrict__ ew,
                       const int*   __restrict__ edges,
                       float*       __restrict__ out) {
  DECLARE_LDS();
  const int tid    = threadIdx.x;
  const int lane   = tid & 31;
  const int nodeLo = blockIdx.x * BIN_NODES;

  for (int i = tid; i < BIN_NODES * FEAT; i += BLOCK) s_acc[i] = 0.0f;
  if (tid < BIN_NODES) s_cnt[tid] = 0.0f;

  stage_tile(0, 0, tid, edges, ew, s_pairs, s_w);
  __syncthreads();

  const float4* __restrict__ feat4 = (const float4*)features;

  for (int t = 0; t < NTILES; ++t) {
    const int cur = t & 1;
    wait_async0();
    __syncthreads();
    if (t + 1 < NTILES)
      stage_tile(1 - cur, (t + 1) * TILE, tid, edges, ew, s_pairs, s_w);
    process_tile(cur, tid, lane, nodeLo, feat4, s_pairs, s_w, s_acc, s_cnt);
  }

  __syncthreads();
  for (int i = tid; i < BIN_NODES * FEAT; i += BLOCK) {
    const int n = i >> 7;
    const int g = nodeLo + n;
    if (g >= N_NODES) continue;
    const int   k   = i & (FEAT - 1);
    const float c0  = s_cnt[n];
    const float c   = fmaxf(c0, EPSV);
    const float msk = (c0 > EPSV) ? 1.0f : 0.0f;
    const float am  = AGG * msk;
    out[g * FEAT + k] = features[g * FEAT + k] * (1.0f - am)
                      + s_acc[i] * (am / c);
  }
}

// ---- Launch --------------------------------------------------------------

extern "C" void kernel_launch(void* const* d_in, const int* in_sizes, int n_in,
                              void* d_out, int out_size, void* d_ws, size_t ws_size,
                              hipStream_t stream) {
  (void)in_sizes; (void)n_in; (void)out_size;
  const float* features = (const float*)d_in[0];
  const float* ew       = (const float*)d_in[1];
  const int*   edges    = (const int*)d_in[2];
  float*       out      = (float*)d_out;

  const size_t ws_need = WS_FLOATS * sizeof(float);
  if (d_ws != nullptr && ws_size >= ws_need) {
    float* ws_acc = (float*)d_ws;
    float* ws_cnt = ws_acc + (size_t)WS_ROWS * FEAT;

    const int nzero = (int)WS_FLOATS;
    zero_ws_kernel<<<(nzero + 255) / 256, 256, 0, stream>>>((float*)d_ws, nzero);

    agg_accum_kernel<<<dim3(NBINS, SLICES), BLOCK, 0, stream>>>(
        features, ew, edges, ws_acc, ws_cnt);

    const int nfin = N_NODES * FEAT;
    finalize_kernel<<<(nfin + 255) / 256, 256, 0, stream>>>(
        features, ws_acc, ws_cnt, out);
  } else {
    agg_single_kernel<<<NBINS, BLOCK, 0, stream>>>(features, ew, edges, out);
  }
}